// GSGNet_25606595019233
// MI455X (gfx1250) — compile-verified
//
#include <hip/hip_runtime.h>
#include <math.h>

#define DIN 64
#define DH  40
#define DOUT 24
#define DH_PAD 48    // layer1 output tiles: 3 x 16
#define DO_PAD 32    // layer2 output tiles: 2 x 16

typedef __attribute__((ext_vector_type(2))) float v2f;
typedef __attribute__((ext_vector_type(8))) float v8f;

__device__ __forceinline__ v8f wmma4(v2f a, v2f b, v8f c) {
  // D = A(16x4 f32) * B(4x16 f32) + C(16x16 f32)
  return __builtin_amdgcn_wmma_f32_16x16x4_f32(false, a, false, b, (short)0, c,
                                               false, false);
}

// ---------------- zero workspace ----------------
__global__ void __launch_bounds__(256)
zero_kernel(float* __restrict__ p, size_t n) {
  size_t i = (size_t)blockIdx.x * blockDim.x + threadIdx.x;
  size_t stride = (size_t)gridDim.x * blockDim.x;
  for (; i < n; i += stride) p[i] = 0.0f;
}

// ---------------- layer-1 edge scatter: x[src] -> sum1[dst], degree count ----
__global__ void __launch_bounds__(256)
scatter1_kernel(const float* __restrict__ x, const int* __restrict__ src,
                const int* __restrict__ dst, float* __restrict__ sum1,
                float* __restrict__ cnt, long E) {
  long t = (long)blockIdx.x * 256 + threadIdx.x;
  long e = t >> 4;                 // 16 lanes per edge, float4 each = 64 floats
  if (e >= E) return;
  int chunk = (int)(t & 15);
  int s = src[e], d = dst[e];
  const float4 v = *reinterpret_cast<const float4*>(x + (size_t)s * DIN + chunk * 4);
  float* dp = sum1 + (size_t)d * DIN + chunk * 4;
  atomicAdd(dp + 0, v.x);
  atomicAdd(dp + 1, v.y);
  atomicAdd(dp + 2, v.z);
  atomicAdd(dp + 3, v.w);
  if (chunk == 0) atomicAdd(cnt + d, 1.0f);
}

// ---------------- layer-2 edge scatter: h[src] -> sum2[dst] -----------------
__global__ void __launch_bounds__(256)
scatter2_kernel(const float* __restrict__ h, const int* __restrict__ src,
                const int* __restrict__ dst, float* __restrict__ sum2, long E) {
  long t = (long)blockIdx.x * 256 + threadIdx.x;
  long e = t / 10;                 // 10 chunks of float4 = 40 floats
  if (e >= E) return;
  int chunk = (int)(t - e * 10);
  int s = src[e], d = dst[e];
  const float4 v = *reinterpret_cast<const float4*>(h + (size_t)s * DH + chunk * 4);
  float* dp = sum2 + (size_t)d * DH + chunk * 4;
  atomicAdd(dp + 0, v.x);
  atomicAdd(dp + 1, v.y);
  atomicAdd(dp + 2, v.z);
  atomicAdd(dp + 3, v.w);
}

// ---------------- layer 1: h = relu(agg@W1l^T + b1 + x@W1r^T) ---------------
// one wave32 per 16-node tile; K=64 in steps of 4; 3 output tiles (N padded 48)
__global__ void __launch_bounds__(256)
layer1_kernel(const float* __restrict__ x, const float* __restrict__ sum1,
              const float* __restrict__ cnt, const float* __restrict__ W1l,
              const float* __restrict__ b1, const float* __restrict__ W1r,
              float* __restrict__ h, int nNodes) {
  __shared__ float sWl[DH_PAD * DIN];   // zero-padded cols 40..47
  __shared__ float sWr[DH_PAD * DIN];
  __shared__ float sB[DH];
  for (int i = threadIdx.x; i < DH_PAD * DIN; i += 256) {
    float wl = (i < DH * DIN) ? W1l[i] : 0.0f;
    float wr = (i < DH * DIN) ? W1r[i] : 0.0f;
    sWl[i] = wl;
    sWr[i] = wr;
  }
  if (threadIdx.x < DH) sB[threadIdx.x] = b1[threadIdx.x];
  __syncthreads();

  int lane = threadIdx.x & 31;
  int wave = threadIdx.x >> 5;
  int node0 = (blockIdx.x * 8 + wave) * 16;
  int row = lane & 15;
  int node = node0 + row;
  int nodeC = min(node, nNodes - 1);   // clamp: OOB rows compute garbage,
                                       // discarded by guarded stores
  int klo = (lane >> 4) * 2;       // A layout: lanes 0-15 hold K=0,1; 16-31 K=2,3
  int ncol = lane & 15;            // B/C column within tile

  float inv = 1.0f / fmaxf(cnt[nodeC], 1.0f);

  const float* xrow = x + (size_t)nodeC * DIN;
  const float* srow = sum1 + (size_t)nodeC * DIN;

  v8f acc0 = {}, acc1 = {}, acc2 = {};
  for (int k = 0; k < DIN; k += 4) {
    v2f aX, aA;
    aX.x = xrow[k + klo];
    aX.y = xrow[k + klo + 1];
    aA.x = srow[k + klo] * inv;
    aA.y = srow[k + klo + 1] * inv;

    v2f bl0, br0, bl1, br1, bl2, br2;
    bl0.x = sWl[ncol * DIN + k + klo];        bl0.y = sWl[ncol * DIN + k + klo + 1];
    br0.x = sWr[ncol * DIN + k + klo];        br0.y = sWr[ncol * DIN + k + klo + 1];
    bl1.x = sWl[(16 + ncol) * DIN + k + klo]; bl1.y = sWl[(16 + ncol) * DIN + k + klo + 1];
    br1.x = sWr[(16 + ncol) * DIN + k + klo]; br1.y = sWr[(16 + ncol) * DIN + k + klo + 1];
    bl2.x = sWl[(32 + ncol) * DIN + k + klo]; bl2.y = sWl[(32 + ncol) * DIN + k + klo + 1];
    br2.x = sWr[(32 + ncol) * DIN + k + klo]; br2.y = sWr[(32 + ncol) * DIN + k + klo + 1];

    acc0 = wmma4(aA, bl0, acc0);
    acc0 = wmma4(aX, br0, acc0);
    acc1 = wmma4(aA, bl1, acc1);
    acc1 = wmma4(aX, br1, acc1);
    acc2 = wmma4(aA, bl2, acc2);
    acc2 = wmma4(aX, br2, acc2);
  }

  int mbase = (lane >> 4) * 8;     // C layout: vgpr r = row r (lanes 0-15) / r+8
  for (int r = 0; r < 8; ++r) {
    int n = node0 + mbase + r;
    if (n < nNodes) {
      float* hrow = h + (size_t)n * DH;
      hrow[ncol]      = fmaxf(acc0[r] + sB[ncol], 0.0f);
      hrow[16 + ncol] = fmaxf(acc1[r] + sB[16 + ncol], 0.0f);
      if (32 + ncol < DH)
        hrow[32 + ncol] = fmaxf(acc2[r] + sB[32 + ncol], 0.0f);
    }
  }
}

// ---------------- layer 2: o = agg2@W2l^T + b2 + h@W2r^T --------------------
// K=40 in steps of 4; 2 output tiles (N padded 32)
__global__ void __launch_bounds__(256)
layer2_kernel(const float* __restrict__ h, const float* __restrict__ sum2,
              const float* __restrict__ cnt, const float* __restrict__ W2l,
              const float* __restrict__ b2, const float* __restrict__ W2r,
              float* __restrict__ o, int nNodes) {
  __shared__ float sWl[DO_PAD * DH];    // zero-padded cols 24..31
  __shared__ float sWr[DO_PAD * DH];
  __shared__ float sB[DOUT];
  for (int i = threadIdx.x; i < DO_PAD * DH; i += 256) {
    float wl = (i < DOUT * DH) ? W2l[i] : 0.0f;
    float wr = (i < DOUT * DH) ? W2r[i] : 0.0f;
    sWl[i] = wl;
    sWr[i] = wr;
  }
  if (threadIdx.x < DOUT) sB[threadIdx.x] = b2[threadIdx.x];
  __syncthreads();

  int lane = threadIdx.x & 31;
  int wave = threadIdx.x >> 5;
  int node0 = (blockIdx.x * 8 + wave) * 16;
  int row = lane & 15;
  int node = node0 + row;
  int nodeC = min(node, nNodes - 1);
  int klo = (lane >> 4) * 2;
  int ncol = lane & 15;

  float inv = 1.0f / fmaxf(cnt[nodeC], 1.0f);

  const float* hrow = h + (size_t)nodeC * DH;
  const float* srow = sum2 + (size_t)nodeC * DH;

  v8f acc0 = {}, acc1 = {};
  for (int k = 0; k < DH; k += 4) {
    v2f aH, aA;
    aH.x = hrow[k + klo];
    aH.y = hrow[k + klo + 1];
    aA.x = srow[k + klo] * inv;
    aA.y = srow[k + klo + 1] * inv;

    v2f bl0, br0, bl1, br1;
    bl0.x = sWl[ncol * DH + k + klo];        bl0.y = sWl[ncol * DH + k + klo + 1];
    br0.x = sWr[ncol * DH + k + klo];        br0.y = sWr[ncol * DH + k + klo + 1];
    bl1.x = sWl[(16 + ncol) * DH + k + klo]; bl1.y = sWl[(16 + ncol) * DH + k + klo + 1];
    br1.x = sWr[(16 + ncol) * DH + k + klo]; br1.y = sWr[(16 + ncol) * DH + k + klo + 1];

    acc0 = wmma4(aA, bl0, acc0);
    acc0 = wmma4(aH, br0, acc0);
    acc1 = wmma4(aA, bl1, acc1);
    acc1 = wmma4(aH, br1, acc1);
  }

  int mbase = (lane >> 4) * 8;
  for (int r = 0; r < 8; ++r) {
    int n = node0 + mbase + r;
    if (n < nNodes) {
      float* orow = o + (size_t)n * DOUT;
      orow[ncol] = acc0[r] + sB[ncol];
      if (16 + ncol < DOUT)
        orow[16 + ncol] = acc1[r] + sB[16 + ncol];
    }
  }
}

// ---------------- log_softmax over 24 classes, one thread per node ----------
__global__ void __launch_bounds__(256)
logsoftmax_kernel(const float* __restrict__ o, float* __restrict__ out, int nNodes) {
  int n = blockIdx.x * 256 + threadIdx.x;
  if (n >= nNodes) return;
  const float* r = o + (size_t)n * DOUT;
  float mx = -3.402823466e+38f;
  for (int j = 0; j < DOUT; ++j) mx = fmaxf(mx, r[j]);
  float s = 0.0f;
  for (int j = 0; j < DOUT; ++j) s += expf(r[j] - mx);
  float lse = mx + logf(s);
  float* w = out + (size_t)n * DOUT;
  for (int j = 0; j < DOUT; ++j) w[j] = r[j] - lse;
}

extern "C" void kernel_launch(void* const* d_in, const int* in_sizes, int n_in,
                              void* d_out, int out_size, void* d_ws, size_t ws_size,
                              hipStream_t stream) {
  const float* x   = (const float*)d_in[0];
  const int*   ei  = (const int*)d_in[1];
  const float* W1l = (const float*)d_in[2];
  const float* b1  = (const float*)d_in[3];
  const float* W1r = (const float*)d_in[4];
  const float* W2l = (const float*)d_in[5];
  const float* b2  = (const float*)d_in[6];
  const float* W2r = (const float*)d_in[7];
  float* out = (float*)d_out;

  int  N = in_sizes[0] / DIN;
  long E = (long)in_sizes[1] / 2;
  const int* src = ei;
  const int* dst = ei + E;

  // workspace layout (floats): sum1 | cnt | sum2 | h | o
  float* ws   = (float*)d_ws;
  float* sum1 = ws;
  float* cnt  = sum1 + (size_t)N * DIN;
  float* sum2 = cnt + N;
  float* h    = sum2 + (size_t)N * DH;
  float* o    = h + (size_t)N * DH;

  size_t nz = (size_t)N * (DIN + 1 + DH);   // sum1 + cnt + sum2
  zero_kernel<<<4096, 256, 0, stream>>>(ws, nz);

  long t1 = E * 16;
  scatter1_kernel<<<(int)((t1 + 255) / 256), 256, 0, stream>>>(x, src, dst, sum1,
                                                               cnt, E);

  int nb = (N + 127) / 128;  // 8 waves/block, 16 nodes/wave
  layer1_kernel<<<nb, 256, 0, stream>>>(x, sum1, cnt, W1l, b1, W1r, h, N);

  long t2 = E * 10;
  scatter2_kernel<<<(int)((t2 + 255) / 256), 256, 0, stream>>>(h, src, dst, sum2, E);

  layer2_kernel<<<nb, 256, 0, stream>>>(h, sum2, cnt, W2l, b2, W2r, o, N);

  logsoftmax_kernel<<<(N + 255) / 256, 256, 0, stream>>>(o, out, N);
}